// Blur_910533066888
// MI455X (gfx1250) — compile-verified
//
#include <hip/hip_runtime.h>

// MI455X / gfx1250 depthwise 4x4 FIR blur, pad=(2,1) both dims.
// Memory-bound (~268 MB @ 23.3 TB/s => ~11.5 us floor). Strategy:
//  - tiny one-thread pre-kernel factors the (flipped) kernel into kv (x) kh and
//    materializes padded banded-coefficient tables (zeros included) in d_ws,
//    so the hot kernel has NO data-dependent selects (no branch trees)
//  - one workgroup (8 wave32) per 64x64 channel-image, staged in LDS
//  - non-temporal streaming float4 global loads/stores (data touched once)
//  - separable FIR as banded matmuls on the matrix pipe:
//      v_wmma_f32_16x16x4_f32, 5 K-chunks per 16x16 tile per pass
//    (fp32 WMMA -> same precision class as the fp32 reference)

typedef float v2f __attribute__((ext_vector_type(2)));
typedef float v4f __attribute__((ext_vector_type(4)));
typedef float v8f __attribute__((ext_vector_type(8)));

#define XPW 76   // Xp row stride: 4 front-pad + 64 data + 8 back-pad (16B-aligned rows)
#define TPW 68   // Tp row stride

// ---- runs once: flip kernel, rank-1 factorize kf = kv (x) kh, emit padded
//      banded tables: tab[s] = kh[s-15] (else 0), tab[40+s] = kv[s-15] (else 0)
__global__ void factor_kernel(const float* __restrict__ ker, float* __restrict__ tab) {
    if (threadIdx.x != 0) return;
    float kf[4][4];
    for (int i = 0; i < 4; ++i)
        for (int j = 0; j < 4; ++j)
            kf[i][j] = ker[(3 - i) * 4 + (3 - j)];
    // pivot column = max abs-sum column
    int jm = 0; float cm = -1.0f;
    for (int j = 0; j < 4; ++j) {
        float s = fabsf(kf[0][j]) + fabsf(kf[1][j]) + fabsf(kf[2][j]) + fabsf(kf[3][j]);
        if (s > cm) { cm = s; jm = j; }
    }
    float kv[4];
    for (int i = 0; i < 4; ++i) kv[i] = kf[i][jm];
    int im = 0; float rm = -1.0f;
    for (int i = 0; i < 4; ++i) {
        float a = fabsf(kv[i]);
        if (a > rm) { rm = a; im = i; }
    }
    float piv = kv[im];
    float kh[4];
    for (int j = 0; j < 4; ++j) kh[j] = (piv != 0.0f) ? kf[im][j] / piv : 0.0f;
    for (int s = 0; s < 40; ++s) {
        int d = s - 15;
        bool in = (d >= 0) && (d < 4);
        tab[s]      = in ? kh[d & 3] : 0.0f;
        tab[40 + s] = in ? kv[d & 3] : 0.0f;
    }
}

__global__ __launch_bounds__(256) void blur_wmma_kernel(
    const float* __restrict__ x, const float* __restrict__ tab,
    float* __restrict__ out)
{
    // Xp: horizontally padded input (logical col c -> storage col c+2; data at cols 4..67),
    //     later reused (cols 0..63) as the output staging tile.
    // Tp: vertically padded intermediate: Tp[r] = Th[r-2], rows 0,1,66,67 = zero.
    // Coef: padded banded coefficients, [0..39] = kh table, [40..79] = kv table.
    __shared__ float Xp[64][XPW];
    __shared__ float Tp[68][TPW];
    __shared__ float Coef[80];

    const int tid  = threadIdx.x;
    const int lane = tid & 31;
    const int wid  = tid >> 5;
    const int mn   = lane & 15;          // M (A/D rows) or N (B/D cols) index
    const int kb   = (lane >> 4) << 1;   // K half-split: lanes 0-15 -> K{0,1}, 16-31 -> K{2,3}
    const long base = (long)blockIdx.x * 4096;

    // ---- zero the pad regions (the zero-coefficient tail chunk still READS
    //      these; WMMA propagates NaN through 0*x, so they must be real zeros) ----
    if (tid < 64) {
        v4f z = {0.f, 0.f, 0.f, 0.f};
        *(v4f*)&Xp[tid][0]  = z;   // logical cols -2,-1 + front slack
        *(v4f*)&Xp[tid][68] = z;   // logical cols 64..  (pad_after + chunk tail)
        *(v4f*)&Xp[tid][72] = z;
    }
    {
        int r = tid >> 6;                    // 0..3
        int row = (r < 2) ? r : (64 + r);    // rows 0,1,66,67
        Tp[row][tid & 63] = 0.0f;
    }
    if (tid < 80) Coef[tid] = tab[tid];      // tiny, L2-broadcast across blocks

    // ---- stream the 64x64 image into LDS (coalesced 512B/wave, non-temporal) ----
    #pragma unroll
    for (int q = 0; q < 4; ++q) {
        int idx4 = tid + (q << 8);
        int row  = idx4 >> 4;
        int col  = (idx4 & 15) << 2;
        v4f v = __builtin_nontemporal_load((const v4f*)(x + base + ((long)idx4 << 2)));
        *(v4f*)&Xp[row][col + 4] = v;
    }
    __syncthreads();

    // ================= pass 1: horizontal banded matmul ==================
    // Th[y][x] = sum_j kh[j] * xpad[y][x+j];  B_t[k][n] = kh[4t+k-n] (banded)
    v2f bb[5];
    #pragma unroll
    for (int t = 0; t < 5; ++t) {
        int ib = 15 + 4 * t + kb - mn;
        bb[t][0] = Coef[ib];
        bb[t][1] = Coef[ib + 1];
    }
    #pragma unroll
    for (int p = 0; p < 2; ++p) {
        int tile = wid + (p << 3);
        int y0 = (tile >> 2) << 4;
        int x0 = (tile & 3) << 4;
        v8f acc = {0.f, 0.f, 0.f, 0.f, 0.f, 0.f, 0.f, 0.f};
        #pragma unroll
        for (int t = 0; t < 5; ++t) {
            int cb = x0 + 4 * t + kb + 2;   // logical col (x0+4t+kb) -> storage col +2
            v2f a;
            a[0] = Xp[y0 + mn][cb];
            a[1] = Xp[y0 + mn][cb + 1];
            acc = __builtin_amdgcn_wmma_f32_16x16x4_f32(
                false, a, false, bb[t], (short)0, acc, false, false);
        }
        int rb = y0 + ((lane >> 4) << 3) + 2;       // Th row + vertical pad offset
        #pragma unroll
        for (int r = 0; r < 8; ++r) Tp[rb + r][x0 + mn] = acc[r];
    }
    __syncthreads();

    // ================= pass 2: vertical banded matmul ====================
    // Out[y][x] = sum_r kv[r-y] * Tp[r][x];  A_t[m][k] = kv[4t+k-m] (banded)
    v2f aa[5];
    #pragma unroll
    for (int t = 0; t < 5; ++t) {
        int ia = 55 + 4 * t + kb - mn;   // 40 (table offset) + 15 (pad offset)
        aa[t][0] = Coef[ia];
        aa[t][1] = Coef[ia + 1];
    }
    #pragma unroll
    for (int p = 0; p < 2; ++p) {
        int tile = wid + (p << 3);
        int y0 = (tile >> 2) << 4;
        int x0 = (tile & 3) << 4;
        v8f acc = {0.f, 0.f, 0.f, 0.f, 0.f, 0.f, 0.f, 0.f};
        #pragma unroll
        for (int t = 0; t < 5; ++t) {
            int rb = y0 + 4 * t + kb;
            v2f b;
            b[0] = Tp[rb][x0 + mn];
            b[1] = Tp[rb + 1][x0 + mn];
            acc = __builtin_amdgcn_wmma_f32_16x16x4_f32(
                false, aa[t], false, b, (short)0, acc, false, false);
        }
        int ro = y0 + ((lane >> 4) << 3);
        #pragma unroll
        for (int r = 0; r < 8; ++r) Xp[ro + r][x0 + mn] = acc[r];  // stage output
    }
    __syncthreads();

    // ---- stream staged output back to global (coalesced, non-temporal) ----
    #pragma unroll
    for (int q = 0; q < 4; ++q) {
        int idx4 = tid + (q << 8);
        int row  = idx4 >> 4;
        int col  = (idx4 & 15) << 2;
        v4f v = *(const v4f*)&Xp[row][col];
        __builtin_nontemporal_store(v, (v4f*)(out + base + ((long)idx4 << 2)));
    }
}

extern "C" void kernel_launch(void* const* d_in, const int* in_sizes, int n_in,
                              void* d_out, int out_size, void* d_ws, size_t ws_size,
                              hipStream_t stream) {
    const float* x   = (const float*)d_in[0];
    const float* ker = (const float*)d_in[1];
    float* out = (float*)d_out;
    float* tab = (float*)d_ws;           // 80 floats: padded kh table, padded kv table
    int n_images = in_sizes[0] / 4096;   // N*C images of 64x64

    factor_kernel<<<dim3(1), dim3(32), 0, stream>>>(ker, tab);
    blur_wmma_kernel<<<dim3(n_images), dim3(256), 0, stream>>>(x, tab, out);
}